// EMDLoss_28999619182610
// MI455X (gfx1250) — compile-verified
//
#include <hip/hip_runtime.h>

typedef __attribute__((ext_vector_type(2))) float v2f;
typedef __attribute__((ext_vector_type(8))) float v8f;

#define N 4096
#define D 32
#define INV_N (1.0f / 4096.0f)

// ---------------------------------------------------------------------------
// Squared row norms of x and y (8192 threads total; tiny kernel)
// ---------------------------------------------------------------------------
__global__ __launch_bounds__(256) void sqnorms_kernel(
    const float* __restrict__ x, const float* __restrict__ y,
    float* __restrict__ x2, float* __restrict__ y2) {
  int t = blockIdx.x * 256 + threadIdx.x;
  const float* src = (t < N) ? (x + (long)t * D) : (y + (long)(t - N) * D);
  float s = 0.f;
#pragma unroll
  for (int k = 0; k < D; ++k) s += src[k] * src[k];
  if (t < N) x2[t] = s; else y2[t - N] = s;
}

// ---------------------------------------------------------------------------
// Build K = exp(-10 * max(0, x2[i] + y2[j] - 2*x_i.y_j)) via f32 WMMA.
// One 16x16 tile per wave; 8 waves per block; grid = 65536/8 = 8192 blocks.
// A frag (16x4 f32): lane<16 -> M=lane, holds {k0,k0+1}; lane>=16 -> {k0+2,k0+3}
// B frag (4x16 f32): mirrored with lane = N; since B = y^T this is the same
// load pattern applied to y's rows.
// C/D (16x16 f32, 8 VGPRs): row = r + 8*(lane>>4), col = lane & 15.
// ---------------------------------------------------------------------------
__global__ __launch_bounds__(256) void build_K_kernel(
    const float* __restrict__ x, const float* __restrict__ y,
    const float* __restrict__ x2, const float* __restrict__ y2,
    float* __restrict__ Km) {
  int wave = threadIdx.x >> 5;
  int lane = threadIdx.x & 31;
  int tile = blockIdx.x * 8 + wave;      // 0 .. 65535
  int ti = tile >> 8;                    // tile row index (0..255)
  int tj = tile & 255;                   // tile col index (0..255)
  int half = lane >> 4;                  // 0 or 1 (selects k-pair)
  int l16  = lane & 15;

  const float* xrow = x + (long)(ti * 16 + l16) * D;  // A: M = l16
  const float* yrow = y + (long)(tj * 16 + l16) * D;  // B: N = l16

  v8f c = {};
#pragma unroll
  for (int k0 = 0; k0 < D; k0 += 4) {
    int kb = k0 + 2 * half;              // even -> 8B aligned
    v2f a = *(const v2f*)(xrow + kb);
    v2f b = *(const v2f*)(yrow + kb);
    c = __builtin_amdgcn_wmma_f32_16x16x4_f32(
        /*neg_a=*/false, a, /*neg_b=*/false, b,
        /*c_mod=*/(short)0, c, /*reuse_a=*/false, /*reuse_b=*/false);
  }

  float yq = y2[tj * 16 + l16];          // column norm for this lane
#pragma unroll
  for (int r = 0; r < 8; ++r) {
    int row = ti * 16 + r + 8 * half;
    float m = x2[row] + yq - 2.0f * c[r];
    m = fmaxf(m, 0.0f);
    Km[(long)row * N + tj * 16 + l16] = __expf(-10.0f * m);
  }
}

// ---------------------------------------------------------------------------
// v <- 1/N (initial marginal scaling)
// ---------------------------------------------------------------------------
__global__ __launch_bounds__(256) void init_v_kernel(float* __restrict__ v) {
  v[blockIdx.x * 256 + threadIdx.x] = INV_N;
}

// ---------------------------------------------------------------------------
// u[i] = (1/N) / (K v)[i].  One wave per row, 8 rows per block, grid = 512.
// v staged in LDS (16 KB); row streamed as float4 (fully coalesced per wave).
// ---------------------------------------------------------------------------
__global__ __launch_bounds__(256) void kv_u_kernel(
    const float* __restrict__ Km, const float* __restrict__ v,
    float* __restrict__ u) {
  __shared__ float vs[N];
  for (int i = threadIdx.x; i < N; i += 256) vs[i] = v[i];
  __syncthreads();

  int wave = threadIdx.x >> 5;
  int lane = threadIdx.x & 31;
  int row = blockIdx.x * 8 + wave;

  const float4* krow = (const float4*)(Km + (long)row * N);
  const float4* vsv  = (const float4*)vs;
  float acc = 0.f;
#pragma unroll 4
  for (int j = lane; j < N / 4; j += 32) {
    float4 k4 = krow[j];
    float4 v4 = vsv[j];
    acc += k4.x * v4.x + k4.y * v4.y + k4.z * v4.z + k4.w * v4.w;
  }
#pragma unroll
  for (int off = 16; off > 0; off >>= 1) acc += __shfl_xor(acc, off, 32);
  if (lane == 0) u[row] = INV_N / acc;
}

// ---------------------------------------------------------------------------
// Partial K^T u: block (bx, by) owns columns [bx*256, bx*256+256) and rows
// [by*128, by*128+128).  All K reads are 1KB coalesced lines.  grid (16, 32).
// ---------------------------------------------------------------------------
__global__ __launch_bounds__(256) void ktu_partial_kernel(
    const float* __restrict__ Km, const float* __restrict__ u,
    float* __restrict__ part) {
  __shared__ float us[128];
  int tid = threadIdx.x;
  int j  = blockIdx.x * 256 + tid;
  int i0 = blockIdx.y * 128;
  if (tid < 128) us[tid] = u[i0 + tid];
  __syncthreads();

  const float* p = Km + (long)i0 * N + j;
  float acc = 0.f;
#pragma unroll 8
  for (int i = 0; i < 128; ++i) acc += p[(long)i * N] * us[i];
  part[(long)blockIdx.y * N + j] = acc;
}

// ---------------------------------------------------------------------------
// v[j] = (1/N) / sum_c part[c][j].  grid = 16.
// ---------------------------------------------------------------------------
__global__ __launch_bounds__(256) void v_update_kernel(
    const float* __restrict__ part, float* __restrict__ v) {
  int j = blockIdx.x * 256 + threadIdx.x;
  float acc = 0.f;
#pragma unroll
  for (int c = 0; c < 32; ++c) acc += part[(long)c * N + j];
  v[j] = INV_N / acc;
}

// ---------------------------------------------------------------------------
// gamma[i][j] = u[i] * K[i][j] * v[j], float4 per thread.  grid = 16384.
// ---------------------------------------------------------------------------
__global__ __launch_bounds__(256) void gamma_kernel(
    const float* __restrict__ Km, const float* __restrict__ u,
    const float* __restrict__ v, float* __restrict__ out) {
  long e = ((long)blockIdx.x * 256 + threadIdx.x) * 4;
  int i = (int)(e >> 12);
  int j = (int)(e & (N - 1));
  float ui = u[i];
  float4 k4 = *(const float4*)(Km + e);
  float4 v4 = *(const float4*)(v + j);
  float4 o;
  o.x = ui * k4.x * v4.x;
  o.y = ui * k4.y * v4.y;
  o.z = ui * k4.z * v4.z;
  o.w = ui * k4.w * v4.w;
  *(float4*)(out + e) = o;
}

// ---------------------------------------------------------------------------
extern "C" void kernel_launch(void* const* d_in, const int* in_sizes, int n_in,
                              void* d_out, int out_size, void* d_ws, size_t ws_size,
                              hipStream_t stream) {
  const float* x = (const float*)d_in[0];
  const float* y = (const float*)d_in[1];
  float* out = (float*)d_out;

  // Workspace layout (floats): K | x2 | y2 | u | v | partials(32*4096)
  float* W    = (float*)d_ws;
  float* Km   = W;
  float* x2   = W + (long)N * N;
  float* y2   = x2 + N;
  float* u    = y2 + N;
  float* v    = u + N;
  float* part = v + N;

  sqnorms_kernel<<<(2 * N) / 256, 256, 0, stream>>>(x, y, x2, y2);
  build_K_kernel<<<(256 * 256) / 8, 256, 0, stream>>>(x, y, x2, y2, Km);
  init_v_kernel<<<N / 256, 256, 0, stream>>>(v);

  for (int it = 0; it < 100; ++it) {
    kv_u_kernel<<<N / 8, 256, 0, stream>>>(Km, v, u);
    ktu_partial_kernel<<<dim3(16, 32), 256, 0, stream>>>(Km, u, part);
    v_update_kernel<<<N / 256, 256, 0, stream>>>(part, v);
  }

  gamma_kernel<<<(long)N * N / 4 / 256, 256, 0, stream>>>(Km, u, v, out);
}